// GraphAttention_52810917872235
// MI455X (gfx1250) — compile-verified
//
#include <hip/hip_runtime.h>

typedef __attribute__((ext_vector_type(2))) float v2f;
typedef __attribute__((ext_vector_type(8))) float v8f;

#define B_BATCH 16
#define N_NODES 2048
#define F_IN    128
#define F_OUT   64
#define ALPHA   0.2f
#define NEG_BIG (-9.0e15f)

// ---------------------------------------------------------------------------
// Kernel 1: Wh = h @ W  (per 16-row tile), plus f1 = Wh@a1, f2 = Wh@a2.
// 128 threads = 4 waves; each wave produces one 16x16 tile of the 16x64 output
// with v_wmma_f32_16x16x4_f32 over K=128.
// ---------------------------------------------------------------------------
__global__ __launch_bounds__(128) void k_wh(
    const float* __restrict__ h, const float* __restrict__ W,
    const float* __restrict__ a,
    float* __restrict__ Wh, float* __restrict__ f1, float* __restrict__ f2)
{
    __shared__ float hs[16][F_IN + 4];      //  8.4 KB
    __shared__ float Ws[F_IN][F_OUT + 4];   // 34.8 KB
    __shared__ float whs[16][F_OUT + 4];    //  4.3 KB

    const int  t    = threadIdx.x;
    const int  wave = t >> 5;
    const int  lane = t & 31;
    const long row0 = (long)blockIdx.x * 16;   // row in [0, B*N)

    // stage h tile: 16 x 128 floats (float4 loads)
    for (int i = t; i < 16 * (F_IN / 4); i += 128) {
        int r = i >> 5, c4 = i & 31;
        float4 v = ((const float4*)(h + (row0 + r) * F_IN))[c4];
        hs[r][c4 * 4 + 0] = v.x; hs[r][c4 * 4 + 1] = v.y;
        hs[r][c4 * 4 + 2] = v.z; hs[r][c4 * 4 + 3] = v.w;
    }
    // stage W: 128 x 64 floats
    for (int i = t; i < F_IN * (F_OUT / 4); i += 128) {
        int r = i >> 4, c4 = i & 15;
        float4 v = ((const float4*)(W + r * F_OUT))[c4];
        Ws[r][c4 * 4 + 0] = v.x; Ws[r][c4 * 4 + 1] = v.y;
        Ws[r][c4 * 4 + 2] = v.z; Ws[r][c4 * 4 + 3] = v.w;
    }
    __syncthreads();

    // WMMA f32 16x16x4: A 16x4 (lane = M row, VGPR g holds K = g + 2*(lane>=16)),
    // B 4x16 (lane%16 = N col), C/D 16x16 f32 in 8 VGPRs.
    const int n0   = wave * 16;
    const int mrow = lane & 15;
    const int kh   = (lane >> 4) << 1;   // 0 or 2
    v8f acc = {};
#pragma unroll
    for (int k = 0; k < F_IN; k += 4) {
        v2f af, bf;
        af.x = hs[mrow][k + kh];
        af.y = hs[mrow][k + kh + 1];
        bf.x = Ws[k + kh][n0 + mrow];
        bf.y = Ws[k + kh + 1][n0 + mrow];
        acc = __builtin_amdgcn_wmma_f32_16x16x4_f32(
            false, af, false, bf, (short)0, acc, false, false);
    }

    // C/D layout: VGPR r -> row (r + 8*(lane>=16)), col = n0 + lane%16
    const int rb = (lane < 16) ? 0 : 8;
#pragma unroll
    for (int rr = 0; rr < 8; ++rr)
        whs[rb + rr][n0 + mrow] = acc[rr];
    __syncthreads();

    // write Wh tile to workspace (float4)
    for (int i = t; i < 16 * (F_OUT / 4); i += 128) {
        int r = i >> 4, c4 = i & 15;
        float4 v = { whs[r][c4 * 4 + 0], whs[r][c4 * 4 + 1],
                     whs[r][c4 * 4 + 2], whs[r][c4 * 4 + 3] };
        ((float4*)(Wh + (row0 + r) * F_OUT))[c4] = v;
    }
    // f1/f2 for the 16 rows
    if (t < 16) {
        float s1 = 0.f, s2 = 0.f;
#pragma unroll 8
        for (int c = 0; c < F_OUT; ++c) {
            float w = whs[t][c];
            s1 += w * a[c];
            s2 += w * a[F_OUT + c];
        }
        f1[row0 + t] = s1;
        f2[row0 + t] = s2;
    }
}

// ---------------------------------------------------------------------------
// Kernel 2: per (batch, 16-row i-tile): masked leaky-relu logits, row softmax,
// write attention, then h_prime tile = attention(16 x 2048) @ Wh[b](2048 x 64).
// 256 threads = 8 waves. GEMM: Wh staged through a 64-row LDS chunk buffer;
// all 8 waves run WMMA (wave&3 = n-tile, wave>>2 = K-half of each chunk),
// partial sums reduced through LDS.
// Dynamic LDS: f2s[2048] + Whs[64][68] + attS[16][2052] = 156,928 B (2/WGP).
// ---------------------------------------------------------------------------
#define LDA   (N_NODES + 4)   // 2052: row stride ≡ 4 mod 64 -> conflict-free A reads
#define LDW   (F_OUT + 4)     // 68
#define CHUNK 64

__global__ __launch_bounds__(256) void k_att(
    const int*   __restrict__ adj, const float* __restrict__ Wh,
    const float* __restrict__ f1,  const float* __restrict__ f2,
    float* __restrict__ hprime, float* __restrict__ attn)
{
    extern __shared__ float smem[];
    float* f2s  = smem;                     // N_NODES floats
    float* Whs  = smem + N_NODES;           // CHUNK * LDW = 4352 floats
    float* attS = Whs + CHUNK * LDW;        // 16 * LDA floats

    const int  t    = threadIdx.x;
    const int  wave = t >> 5;
    const int  lane = t & 31;
    const int  b    = blockIdx.x >> 7;     // N/16 = 128 tiles per batch
    const int  i0   = (blockIdx.x & 127) * 16;
    const long rowbase = (long)b * N_NODES + i0;

    // stage f2[b, :]
    const float4* f2g = (const float4*)(f2 + (long)b * N_NODES);
    for (int i = t; i < N_NODES / 4; i += 256) ((float4*)f2s)[i] = f2g[i];
    __syncthreads();

    // 16 lanes per row; rows 2w, 2w+1 live in wave w -> shuffle width 16 is in-wave
    const int   r    = t >> 4;
    const int   c    = t & 15;
    const float f1v  = f1[rowbase + r];
    const int*  adjR = adj + (rowbase + r) * (long)N_NODES;
    float*      arow = attS + r * LDA;

    // pass A: masked leaky-relu logits + running max
    float lmax = NEG_BIG;
    for (int q = c; q < N_NODES / 4; q += 16) {
        __builtin_prefetch(((const int4*)adjR) + q + 16, 0, 0);
        int4   av = ((const int4*)adjR)[q];
        float4 fv = ((const float4*)f2s)[q];
        float e0 = f1v + fv.x; e0 = e0 > 0.f ? e0 : ALPHA * e0; e0 = av.x > 0 ? e0 : NEG_BIG;
        float e1 = f1v + fv.y; e1 = e1 > 0.f ? e1 : ALPHA * e1; e1 = av.y > 0 ? e1 : NEG_BIG;
        float e2 = f1v + fv.z; e2 = e2 > 0.f ? e2 : ALPHA * e2; e2 = av.z > 0 ? e2 : NEG_BIG;
        float e3 = f1v + fv.w; e3 = e3 > 0.f ? e3 : ALPHA * e3; e3 = av.w > 0 ? e3 : NEG_BIG;
        float4 o = { e0, e1, e2, e3 };
        ((float4*)arow)[q] = o;
        lmax = fmaxf(lmax, fmaxf(fmaxf(e0, e1), fmaxf(e2, e3)));
    }
    for (int m = 8; m; m >>= 1) lmax = fmaxf(lmax, __shfl_xor(lmax, m, 16));

    // pass B: exp + sum  (all-masked row -> exp(0)=1 everywhere -> uniform, as JAX)
    float lsum = 0.f;
    for (int q = c; q < N_NODES / 4; q += 16) {
        float4 v = ((float4*)arow)[q];
        v.x = __expf(v.x - lmax); v.y = __expf(v.y - lmax);
        v.z = __expf(v.z - lmax); v.w = __expf(v.w - lmax);
        ((float4*)arow)[q] = v;
        lsum += (v.x + v.y) + (v.z + v.w);
    }
    for (int m = 8; m; m >>= 1) lsum += __shfl_xor(lsum, m, 16);
    const float inv = 1.0f / lsum;

    // pass C: normalize; write attention row to global + keep in LDS for GEMM
    float* aout = attn + (rowbase + r) * (long)N_NODES;
    for (int q = c; q < N_NODES / 4; q += 16) {
        float4 v = ((float4*)arow)[q];
        v.x *= inv; v.y *= inv; v.z *= inv; v.w *= inv;
        ((float4*)arow)[q] = v;
        ((float4*)aout)[q] = v;
    }
    __syncthreads();

    // ---- GEMM: h_prime tile = attS(16 x 2048) @ Wh[b](2048 x 64) ----
    // wave&3 -> 16-col n-tile; wave>>2 -> 32-row K-half of each 64-row chunk.
    const int n0   = (wave & 3) * 16;
    const int ksub = (wave >> 2) * 32;
    const int mrow = lane & 15;
    const int kh   = (lane >> 4) << 1;
    const float* WhB = Wh + (long)b * N_NODES * F_OUT;

    v8f acc = {};
    for (int c0 = 0; c0 < N_NODES; c0 += CHUNK) {
        // cooperative stage: CHUNK rows x 64 cols of Wh[b] (float4, coalesced)
        for (int i = t; i < CHUNK * (F_OUT / 4); i += 256) {
            int rr = i >> 4, c4 = i & 15;
            float4 v = ((const float4*)(WhB + (long)(c0 + rr) * F_OUT))[c4];
            Whs[rr * LDW + c4 * 4 + 0] = v.x;
            Whs[rr * LDW + c4 * 4 + 1] = v.y;
            Whs[rr * LDW + c4 * 4 + 2] = v.z;
            Whs[rr * LDW + c4 * 4 + 3] = v.w;
        }
        __syncthreads();
#pragma unroll
        for (int jj = 0; jj < 32; jj += 4) {
            const int j = ksub + jj;
            v2f af, bf;
            af.x = attS[mrow * LDA + c0 + j + kh];
            af.y = attS[mrow * LDA + c0 + j + kh + 1];
            bf.x = Whs[(j + kh) * LDW + n0 + mrow];
            bf.y = Whs[(j + kh + 1) * LDW + n0 + mrow];
            acc = __builtin_amdgcn_wmma_f32_16x16x4_f32(
                false, af, false, bf, (short)0, acc, false, false);
        }
        __syncthreads();
    }

    // cross-wave K reduction: waves 4-7 dump partials (reuse Whs region: 4 KB)
    float* red = Whs;
    if (wave >= 4) {
#pragma unroll
        for (int rr = 0; rr < 8; ++rr)
            red[((wave - 4) * 8 + rr) * 32 + lane] = acc[rr];
    }
    __syncthreads();
    if (wave < 4) {
        const int rb = (lane < 16) ? 0 : 8;
#pragma unroll
        for (int rr = 0; rr < 8; ++rr) {
            float v = acc[rr] + red[(wave * 8 + rr) * 32 + lane];
            hprime[(rowbase + rb + rr) * F_OUT + n0 + mrow] = v;
        }
    }
}

// ---------------------------------------------------------------------------
extern "C" void kernel_launch(void* const* d_in, const int* in_sizes, int n_in,
                              void* d_out, int out_size, void* d_ws, size_t ws_size,
                              hipStream_t stream)
{
    const float* h   = (const float*)d_in[0];
    const int*   adj = (const int*)  d_in[1];
    const float* W   = (const float*)d_in[2];
    const float* a   = (const float*)d_in[3];

    // workspace: Wh [B*N*64] | f1 [B*N] | f2 [B*N]  (~8.5 MB)
    float* Wh = (float*)d_ws;
    float* f1 = Wh + (long)B_BATCH * N_NODES * F_OUT;
    float* f2 = f1 + (long)B_BATCH * N_NODES;

    // outputs: h_prime [B*N*64] then attention [B*N*N]
    float* hprime = (float*)d_out;
    float* attn   = hprime + (long)B_BATCH * N_NODES * F_OUT;

    k_wh<<<B_BATCH * N_NODES / 16, 128, 0, stream>>>(h, W, a, Wh, f1, f2);

    const size_t smem = (size_t)(N_NODES + CHUNK * LDW + 16 * LDA) * sizeof(float);
    hipFuncSetAttribute(reinterpret_cast<const void*>(k_att),
                        hipFuncAttributeMaxDynamicSharedMemorySize, (int)smem);
    k_att<<<B_BATCH * (N_NODES / 16), 256, smem, stream>>>(adj, Wh, f1, f2, hprime, attn);
}